// LMFEAM_SSN_54606214202177
// MI455X (gfx1250) — compile-verified
//
#include <hip/hip_runtime.h>
#include <math.h>

// ---------------------------------------------------------------------------
// Fused soft-SLIC for MI455X (gfx1250, wave32).
// f32 WMMA 16x16x4 for both GEMMs; Q never touches HBM except final output.
// ---------------------------------------------------------------------------

typedef __attribute__((ext_vector_type(2))) float v2f;
typedef __attribute__((ext_vector_type(8))) float v8f;

#define NSPIX   512
#define NPIX    32768
#define NCH     20
#define NCHE    21                 // channels + ones row (for den)
#define WAVES   2                  // waves per workgroup
#define PXW     16                 // pixels per wave (WMMA tile width)
#define CHUNK   (WAVES*PXW)        // 32 pixels per chunk
#define NCHUNKS 4                  // chunks per workgroup
#define NWG     256                // 256*4*32 = 32768 pixels
#define TPB     (WAVES*32)         // 64 threads
#define DSTR    513                // dist-tile LDS row stride (conflict-free)
#define FSTR    25                 // feature-tile LDS row stride (conflict-free)
#define NUMSTR  22                 // num LDS/global row stride (c = 0..21)
#define SLIC_EPS 1e-16f

__global__ __launch_bounds__(TPB, 1)
void slic_iter(const float* __restrict__ feat,   // [20][32768] c-major
               const float* __restrict__ cT,     // [20][512]   c-major
               float* __restrict__ numG,         // [512][22] atomic accumulator (pre-zeroed)
               float* __restrict__ qOut,         // [512][32768] (final iter only)
               float* __restrict__ labOut,       // [32768]      (final iter only)
               int writeOut)
{
    __shared__ float sC[NCH*NSPIX];     // centers [k][s]
    __shared__ float sCn[NSPIX];        // ||c_s||^2
    __shared__ float sF[CHUNK*FSTR];    // feature tile [pixel][c], c=20 is ones
    __shared__ float sT[CHUNK*DSTR];    // t / e values [pixel][s]
    __shared__ float sNum[NSPIX*NUMSTR];// per-WG num accumulator [s][c]
    __shared__ float sPM[TPB];          // partial max
    __shared__ int   sPA[TPB];          // partial argmax
    __shared__ float sPS[TPB];          // partial sum
    __shared__ float sM[CHUNK];         // per-pixel max
    __shared__ float sR[CHUNK];         // per-pixel 1/den
    __shared__ int   sArg[CHUNK];       // per-pixel argmax (labels)

    const int tid  = threadIdx.x;
    const int wave = tid >> 5;
    const int lane = tid & 31;
    const int hl   = lane & 15;   // lane within half-wave
    const int hi   = lane >> 4;   // 0: lanes 0-15, 1: lanes 16-31

    // Load centers, zero num accumulator
    for (int i = tid; i < NCH*NSPIX; i += TPB) sC[i] = cT[i];
    for (int i = tid; i < NSPIX*NUMSTR; i += TPB) sNum[i] = 0.f;
    __syncthreads();
    // ||c_s||^2
    for (int s = tid; s < NSPIX; s += TPB) {
        float a = 0.f;
        for (int k = 0; k < NCH; ++k) { float v = sC[k*NSPIX + s]; a += v*v; }
        sCn[s] = a;
    }
    __syncthreads();

    for (int chunk = 0; chunk < NCHUNKS; ++chunk) {
        const int nb = (blockIdx.x * NCHUNKS + chunk) * CHUNK;

        // ---- Stage feature tile (transposed into [pixel][c]) + ones row ----
        for (int i = tid; i < CHUNK*NCH; i += TPB) {
            int c = i / CHUNK, p = i % CHUNK;       // lanes: consecutive p -> coalesced
            sF[p*FSTR + c] = feat[c*NPIX + nb + p];
        }
        for (int p = tid; p < CHUNK; p += TPB) sF[p*FSTR + NCH] = 1.f;
        __syncthreads();

        // ---- Pass 1: t[s,n] = 2*(c_s . f_n) - ||c_s||^2 via WMMA f32 16x16x4
        {
            const int pb = wave * PXW;
            v2f bfrag[5];
            for (int kc = 0; kc < 5; ++kc) {
                // B (K=4 x N=16): lanes 0-15 -> K=0,1 ; lanes 16-31 -> K=2,3
                int k0 = kc*4 + hi*2;
                int p  = pb + hl;
                v2f b; b.x = sF[p*FSTR + k0]; b.y = sF[p*FSTR + k0 + 1];
                bfrag[kc] = b;
            }
            for (int sb = 0; sb < NSPIX/16; ++sb) {
                int s = sb*16 + hl;
                v8f acc = {};
                for (int kc = 0; kc < 5; ++kc) {
                    // A (M=16 x K=4): lanes 0-15 rows, K=0,1 ; lanes 16-31 rows, K=2,3
                    int k0 = kc*4 + hi*2;
                    v2f a; a.x = sC[k0*NSPIX + s]; a.y = sC[(k0+1)*NSPIX + s];
                    acc = __builtin_amdgcn_wmma_f32_16x16x4_f32(
                            false, a, false, bfrag[kc], (short)0, acc, false, false);
                }
                // D tile: vgpr i, lanes 0-15: s=sb*16+i ; lanes 16-31: s=sb*16+i+8
                int prow = pb + hl;
                #pragma unroll
                for (int i = 0; i < 8; ++i) {
                    int srow = sb*16 + i + hi*8;
                    sT[prow*DSTR + srow] = 2.f*acc[i] - sCn[srow];
                }
            }
        }
        __syncthreads();

        // ---- Softmax over s (2 threads per pixel); e = exp(t - m) in place
        {
            int p = tid >> 1, half = tid & 1;
            int s0 = half * (NSPIX/2);
            float m = -3.0e38f; int am = 0;
            for (int s = s0; s < s0 + NSPIX/2; ++s) {
                float v = sT[p*DSTR + s];
                if (v > m) { m = v; am = s; }
            }
            sPM[tid] = m; sPA[tid] = am;
            __syncthreads();
            if (half == 0) {
                float m0 = sPM[tid], m1 = sPM[tid+1];
                sM[p]   = (m1 > m0) ? m1 : m0;
                sArg[p] = (m1 > m0) ? sPA[tid+1] : sPA[tid];
            }
            __syncthreads();
            float mm = sM[p];
            float sum = 0.f;
            for (int s = s0; s < s0 + NSPIX/2; ++s) {
                float e = __expf(sT[p*DSTR + s] - mm);
                sT[p*DSTR + s] = e;
                sum += e;
            }
            sPS[tid] = sum;
            __syncthreads();
            if (half == 0) sR[p] = 1.f / (sPS[tid] + sPS[tid+1]);
        }
        __syncthreads();

        // ---- Pass 2: num[s][c] += sum_n e[s,n] * (r_n * fext[c,n]) via WMMA
        // den folded in as channel 20 (ones row), 1/den folded into B columns.
        {
            const int pb = wave * PXW;
            for (int ct = 0; ct < 2; ++ct) {
                int c = ct*16 + hl;                 // output channel for this lane
                v2f bfrag[4];
                for (int nc = 0; nc < 4; ++nc) {
                    int p0 = nc*4 + hi*2;           // pixel rows K
                    float r0 = sR[pb + p0], r1 = sR[pb + p0 + 1];
                    v2f b;
                    b.x = (c < NCHE) ? sF[(pb+p0  )*FSTR + c] * r0 : 0.f;
                    b.y = (c < NCHE) ? sF[(pb+p0+1)*FSTR + c] * r1 : 0.f;
                    bfrag[nc] = b;
                }
                for (int sb = 0; sb < NSPIX/16; ++sb) {
                    int s = sb*16 + hl;
                    v8f acc = {};
                    for (int nc = 0; nc < 4; ++nc) {
                        int p0 = nc*4 + hi*2;
                        v2f a;
                        a.x = sT[(pb+p0  )*DSTR + s];
                        a.y = sT[(pb+p0+1)*DSTR + s];
                        acc = __builtin_amdgcn_wmma_f32_16x16x4_f32(
                                false, a, false, bfrag[nc], (short)0, acc, false, false);
                    }
                    #pragma unroll
                    for (int i = 0; i < 8; ++i) {
                        int srow = sb*16 + i + hi*8;
                        if (c < NUMSTR)
                            unsafeAtomicAdd(&sNum[srow*NUMSTR + c], acc[i]);
                    }
                }
            }
        }

        // ---- Final-iteration outputs: Q (coalesced) and labels
        if (writeOut) {
            for (int i = tid; i < NSPIX*CHUNK; i += TPB) {
                int s = i / CHUNK, p = i % CHUNK;   // consecutive lanes -> consecutive n
                qOut[s*NPIX + nb + p] = sT[p*DSTR + s] * sR[p];
            }
            for (int p = tid; p < CHUNK; p += TPB)
                labOut[nb + p] = (float)sArg[p];
        }
        __syncthreads();
    }

    // ---- Flush per-WG num accumulator to global
    for (int i = tid; i < NSPIX*NUMSTR; i += TPB)
        unsafeAtomicAdd(&numG[i], sNum[i]);
}

__global__ void slic_init(const float* __restrict__ feat, float* __restrict__ cT)
{
    int i = blockIdx.x * blockDim.x + threadIdx.x;
    if (i < NCH*NSPIX) {
        int c = i / NSPIX, s = i % NSPIX;
        cT[i] = feat[c*NPIX + s];       // centers0 = features[:, :, :512]
    }
}

__global__ void slic_zero(float* __restrict__ numG)
{
    int i = blockIdx.x * blockDim.x + threadIdx.x;
    if (i < NSPIX*NUMSTR) numG[i] = 0.f;
}

__global__ void slic_update(const float* __restrict__ numG, float* __restrict__ cT,
                            float* __restrict__ centOut, int writeOut)
{
    int i = blockIdx.x * blockDim.x + threadIdx.x;
    if (i < NCH*NSPIX) {
        int c = i / NSPIX, s = i % NSPIX;
        float den = numG[s*NUMSTR + NCH] + SLIC_EPS;
        float v = numG[s*NUMSTR + c] / den;
        cT[i] = v;
        if (writeOut) centOut[i] = v;   // output layout (B,C,S) = c*512+s
    }
}

extern "C" void kernel_launch(void* const* d_in, const int* in_sizes, int n_in,
                              void* d_out, int out_size, void* d_ws, size_t ws_size,
                              hipStream_t stream)
{
    const float* feat = (const float*)d_in[0];

    // Output tuple (Q, labels, centers) concatenated flat
    float* qOut    = (float*)d_out;                  // 512*32768
    float* labOut  = qOut + (size_t)NSPIX*NPIX;      // 32768 (argmax as float)
    float* centOut = labOut + NPIX;                  // 20*512

    // Workspace: centers [20][512] + num [512][22]
    float* cT   = (float*)d_ws;
    float* numG = cT + NCH*NSPIX;

    slic_init<<<(NCH*NSPIX + 255)/256, 256, 0, stream>>>(feat, cT);
    for (int it = 0; it < 10; ++it) {
        int wo = (it == 9) ? 1 : 0;
        slic_zero<<<(NSPIX*NUMSTR + 255)/256, 256, 0, stream>>>(numG);
        slic_iter<<<NWG, TPB, 0, stream>>>(feat, cT, numG, qOut, labOut, wo);
        slic_update<<<(NCH*NSPIX + 255)/256, 256, 0, stream>>>(numG, cT, centOut, wo);
    }
}